// MinArchitecture_19585050870361
// MI455X (gfx1250) — compile-verified
//
#include <hip/hip_runtime.h>

// ---------------- problem constants (from reference) ----------------
#define BATCH 65536
#define SEQ   512

// ---------------- tiling ----------------
#define RPB   256                // rows per block (= threads per block, 8 waves)
#define CH    16                 // sequence columns per chunk
#define NCH   (SEQ / CH)         // 32 chunks
#define STRIDE 18                // padded LDS row stride (floats): conflict-free for wave32, 8B-aligned
#define NBUF  3                  // triple buffer -> single barrier per chunk
#define OPS_PER_CHUNK 8          // async b64 copies issued per thread per chunk (256*16 floats / 2 / 256)

typedef int v2i __attribute__((ext_vector_type(2)));
typedef __attribute__((address_space(1))) v2i  g_v2i;   // global (prints as __device__)
typedef __attribute__((address_space(3))) v2i  l_v2i;   // LDS
typedef __attribute__((address_space(3))) char lchar;

// one 8-byte async global->LDS copy (tracked by ASYNCcnt)
__device__ __forceinline__ void async_copy_b64(const float* g, float* l) {
#if __has_builtin(__builtin_amdgcn_global_load_async_to_lds_b64)
    __builtin_amdgcn_global_load_async_to_lds_b64((g_v2i*)g, (l_v2i*)l, 0, 0);
#else
    unsigned lds_off = (unsigned)(unsigned long long)(lchar*)l;
    asm volatile("global_load_async_to_lds_b64 %0, %1, off"
                 :: "v"(lds_off), "v"(g) : "memory");
#endif
}

template <int N>
__device__ __forceinline__ void wait_asynccnt_le() {
#if __has_builtin(__builtin_amdgcn_s_wait_asynccnt)
    __builtin_amdgcn_s_wait_asynccnt((unsigned short)N);
#else
    asm volatile("s_wait_asynccnt %0" :: "i"(N) : "memory");
#endif
}

__device__ __forceinline__ float fast_rcp(float x) {
    return __builtin_amdgcn_rcpf(x);
}
__device__ __forceinline__ float fast_exp(float x) {   // e^x
    return __builtin_amdgcn_exp2f(x * 1.4426950408889634f);
}
// native v_tanh_f32 on gfx1250 if the builtin exists; exp2-based fallback otherwise
__device__ __forceinline__ float fast_tanh(float x) {
#if __has_builtin(__builtin_amdgcn_tanhf)
    return __builtin_amdgcn_tanhf(x);
#elif __has_builtin(__builtin_amdgcn_tanh_f32)
    return __builtin_amdgcn_tanh_f32(x);
#else
    const float e = fast_exp(2.0f * x);
    return 1.0f - 2.0f * fast_rcp(e + 1.0f);
#endif
}
__device__ __forceinline__ float fast_sigmoid(float x) {
    // sigmoid(x) = 0.5 + 0.5*tanh(x/2)  -> one TRANS op + one FMA
    return __builtin_fmaf(0.5f, fast_tanh(0.5f * x), 0.5f);
}

__global__ __launch_bounds__(RPB)
void min_arch_scan_kernel(const float* __restrict__ X,
                          const float* __restrict__ W1,
                          const float* __restrict__ W2,
                          const float* __restrict__ bias,
                          const float* __restrict__ Wz,
                          const float* __restrict__ Ws,
                          float* __restrict__ out)
{
    __shared__ float tile[NBUF][RPB * STRIDE];   // 3 x 18KB rotating buffers

    const int tid     = threadIdx.x;
    const int rowBase = blockIdx.x * RPB;

    // ---- scalar parameters (uniform; softmax over 2-vector == sigmoid of diff) ----
    const float a  = fast_sigmoid(W1[0] - W1[1]) - fast_sigmoid(W2[0] - W2[1]);
    const float b0 = bias[0];
    const float wz = Wz[0];
    const float ws = Ws[0];

    // ---- issue async load of chunk 0 into buffer 0 ----
    {
        #pragma unroll
        for (int k = 0; k < OPS_PER_CHUNK; ++k) {
            const int idx = tid + k * RPB;          // 0..2047  (b64 elements in tile)
            const int r   = idx >> 3;               // row in tile  (8 x b64 per row)
            const int cp  = idx & 7;                // column pair
            const float* g = X + (size_t)(rowBase + r) * SEQ + (size_t)(cp * 2);
            async_copy_b64(g, &tile[0][r * STRIDE + cp * 2]);
        }
    }

    float h = 0.0f;
    int cur = 0;                 // buffer holding chunk c
    int nxt = 1;                 // buffer receiving chunk c+1

    for (int c = 0; c < NCH; ++c) {
        if (c + 1 < NCH) {
            // Prefetch chunk c+1 into buffer nxt. Overwrite hazard is with compute
            // of chunk c-2, already protected by the barrier of iteration c-1.
            #pragma unroll
            for (int k = 0; k < OPS_PER_CHUNK; ++k) {
                const int idx = tid + k * RPB;
                const int r   = idx >> 3;
                const int cp  = idx & 7;
                const float* g = X + (size_t)(rowBase + r) * SEQ
                                   + (size_t)(c + 1) * CH + (size_t)(cp * 2);
                async_copy_b64(g, &tile[nxt][r * STRIDE + cp * 2]);
            }
            // async loads complete in order: <=8 outstanding means chunk c is in LDS.
            wait_asynccnt_le<OPS_PER_CHUNK>();
        } else {
            wait_asynccnt_le<0>();
        }
        __syncthreads();          // all waves' chunk-c data visible (single barrier/chunk)

        // ---- 16 recurrence steps from my LDS row (bank-conflict-free: stride 18) ----
        const float* mr = &tile[cur][tid * STRIDE];
        #pragma unroll
        for (int j = 0; j < CH; ++j) {
            const float x = mr[j];
            if (c == 0 && j == 0) { h = x; continue; }   // h0 = X[:,0]
            const float d = h - x;
            const float t = fast_tanh(a * d);            // ZETA = 1
            const float u = b0 + ws * t + wz * (1.0f - 2.0f * t * t);
            const float s = fast_sigmoid(u);
            h = x + s * d;                               // s*h + (1-s)*x
        }

        // rotate buffers: cur <- nxt <- third
        const int third = 3 - cur - nxt;   // {0,1,2} \ {cur,nxt}
        cur = nxt;
        nxt = third;
    }

    out[rowBase + tid] = h;
}

extern "C" void kernel_launch(void* const* d_in, const int* in_sizes, int n_in,
                              void* d_out, int out_size, void* d_ws, size_t ws_size,
                              hipStream_t stream) {
    (void)in_sizes; (void)n_in; (void)d_ws; (void)ws_size; (void)out_size;
    const float* X    = (const float*)d_in[0];
    const float* W1   = (const float*)d_in[1];
    const float* W2   = (const float*)d_in[2];
    const float* bias = (const float*)d_in[3];
    const float* Wz   = (const float*)d_in[4];
    const float* Ws   = (const float*)d_in[5];
    float*       out  = (float*)d_out;

    dim3 grid(BATCH / RPB);     // 256 blocks
    dim3 block(RPB);            // 256 threads = 8 wave32
    min_arch_scan_kernel<<<grid, block, 0, stream>>>(X, W1, W2, bias, Wz, Ws, out);
}